// MDGNN_65000035058013
// MI455X (gfx1250) — compile-verified
//
#include <hip/hip_runtime.h>
#include <hip/hip_bf16.h>

typedef float v2f __attribute__((ext_vector_type(2)));
typedef float v8f __attribute__((ext_vector_type(8)));

#define D 128
#define DE 16
#define LN_EPS 1e-5f

// ---------- monotone float<->uint encoding for atomic segment-max ----------
__device__ __forceinline__ unsigned int encode_f32(float f) {
  unsigned int u = __float_as_uint(f);
  return (u & 0x80000000u) ? ~u : (u | 0x80000000u);
}
__device__ __forceinline__ float decode_f32(unsigned int u) {
  unsigned int v = (u & 0x80000000u) ? (u & 0x7FFFFFFFu) : ~u;
  return __uint_as_float(v);
}

// ---------------------------------------------------------------------------
// K-pair-interleaved, bank-skewed LDS layout for the 128x128 weight matrix.
// Element (k, col) lives at:  (k/2)*256 + ((2*col + ((k/2)&1)*32) & 255) + (k&1)
//  - the pair {(2kp,col),(2kp+1,col)} is contiguous -> one aligned ds_load_b64
//    directly into an even VGPR pair (WMMA SRC1 needs even alignment, so this
//    kills the v_dual_mov shuffles the row-major layout caused)
//  - odd-kp rows are rotated by 32 floats so the two 16-lane halves of a wave
//    (which read kp and kp+1) land on disjoint 32-bank halves: a wave32
//    ds_load_b64 becomes a conflict-free full-width (64-bank) LDS access.
// ---------------------------------------------------------------------------
__device__ __forceinline__ int wp_index(int k, int col) {
  const int kp = k >> 1;
  return kp * 256 + (((col << 1) + ((kp & 1) << 5)) & 255) + (k & 1);
}

// ---------------------------------------------------------------------------
// One wave computes C[16,128] = X[16-row tile, 128] @ W[128,128] with
// V_WMMA_F32_16X16X4_F32.  W resides in LDS (paired layout above).
// WMMA layouts per CDNA5 ISA 7.12.2:
//   A 16x4 f32: lanes 0-15 M=lane {K=k0,k0+1}; lanes 16-31 {K=k0+2,k0+3}
//   B 4x16 f32: VGPR v: lanes 0-15 K=k0+v, lanes 16-31 K=k0+2+v; N=lane%16
//   C/D 16x16:  VGPR v: lanes 0-15 row v, lanes 16-31 row v+8; col=n0+lane%16
// ---------------------------------------------------------------------------
__device__ __forceinline__ void wmma_row_gemm(const float* __restrict__ X, int mC,
                                              const float* __restrict__ Wp,
                                              int hi, int lo, v8f* acc) {
  const float* xr = X + (size_t)mC * D;
  const int skew = hi << 5;      // kp parity == hi (k0 is a multiple of 4)
  const int colb = lo << 1;
  for (int k0 = 0; k0 < D; k0 += 4) {
    // A fragment: contiguous pair -> global_load_b64
    const v2f a = *(const v2f*)&xr[k0 + 2 * hi];
    const float* wrow = Wp + ((k0 >> 1) + hi) * 256;
#pragma unroll
    for (int nt = 0; nt < 8; ++nt) {
      // B fragment: one aligned ds_load_b64 into an even VGPR pair
      const v2f b = *(const v2f*)&wrow[((nt << 5) + colb + skew) & 255];
      acc[nt] = __builtin_amdgcn_wmma_f32_16x16x4_f32(
          false, a, false, b, (short)0, acc[nt], false, false);
    }
  }
}

__device__ __forceinline__ void zero_acc(v8f* acc) {
#pragma unroll
  for (int nt = 0; nt < 8; ++nt)
#pragma unroll
    for (int i = 0; i < 8; ++i) acc[nt][i] = 0.0f;
}

// cooperative fill of the paired/skewed LDS image (coalesced global reads)
__device__ __forceinline__ void fill_wlds(float* __restrict__ Wp,
                                          const float* __restrict__ W,
                                          int tid, int nthreads) {
  for (int i = tid; i < D * D; i += nthreads) {
    const int k = i >> 7;
    const int col = i & 127;
    Wp[wp_index(k, col)] = W[i];
  }
}

// ---------------------------------------------------------------------------
// Stage 0: fuse weights.
//   Wsm[D,D]  = W_src @ W_msg
//   Wem[DE,D] = W_edge @ W_msg
//   bmt[D]    = (b_src + b_edge) @ W_msg + b_msg
// ---------------------------------------------------------------------------
__global__ __launch_bounds__(D) void fuse_weights_kernel(
    const float* __restrict__ W_src, const float* __restrict__ W_msg,
    const float* __restrict__ W_edge, const float* __restrict__ b_src,
    const float* __restrict__ b_edge, const float* __restrict__ b_msg,
    float* __restrict__ Wsm, float* __restrict__ Wem, float* __restrict__ bmt) {
  const int r = blockIdx.x;
  const int d = threadIdx.x;
  if (r < D) {
    float acc = 0.f;
    for (int k = 0; k < D; ++k) acc += W_src[r * D + k] * W_msg[k * D + d];
    Wsm[r * D + d] = acc;
  } else if (r < D + DE) {
    const int rr = r - D;
    float acc = 0.f;
    for (int k = 0; k < D; ++k) acc += W_edge[rr * D + k] * W_msg[k * D + d];
    Wem[rr * D + d] = acc;
  } else {
    float acc = b_msg[d];
    for (int k = 0; k < D; ++k) acc += (b_src[k] + b_edge[k]) * W_msg[k * D + d];
    bmt[d] = acc;
  }
}

// ---------------------------------------------------------------------------
// Stage 1/2: per-node.  Pass A: F = X@W1 + b1 -> s_out[n] = dot(tanh(F), Wa).
//            Pass B: P_out = X@W2 (raw, biases folded elsewhere).
// 128 threads = 4 waves; each wave owns an independent 16-row tile.
// ---------------------------------------------------------------------------
__global__ __launch_bounds__(128) void node_stage_kernel(
    const float* __restrict__ X, int N, const float* __restrict__ W1,
    const float* __restrict__ b1, const float* __restrict__ Wa,
    float* __restrict__ s_out, const float* __restrict__ W2,
    float* __restrict__ P_out) {
  __shared__ float Wlds[D * D];
  const int tid = threadIdx.x;
  const int wave = tid >> 5, lane = tid & 31, lo = lane & 15, hi = lane >> 4;
  const int rowBase = blockIdx.x * 64 + wave * 16;
  const bool active = rowBase < N;
  const int m = rowBase + lo;
  const int mC = (m < N) ? m : (N - 1);

  fill_wlds(Wlds, W1, tid, 128);
  __syncthreads();

  v8f acc[8];
  if (active) {
    zero_acc(acc);
    wmma_row_gemm(X, mC, Wlds, hi, lo, acc);
    // attention scalar: per lane, partial sums for rows (v + 8*hi)
    float sv[8];
#pragma unroll
    for (int v = 0; v < 8; ++v) sv[v] = 0.f;
#pragma unroll
    for (int nt = 0; nt < 8; ++nt) {
      const int col = nt * 16 + lo;
      const float wac = Wa[col];
      const float bc = b1[col];
#pragma unroll
      for (int v = 0; v < 8; ++v) sv[v] += tanhf(acc[nt][v] + bc) * wac;
    }
#pragma unroll
    for (int v = 0; v < 8; ++v) {  // butterfly over the 16-lane half-wave
      sv[v] += __shfl_xor(sv[v], 1, 32);
      sv[v] += __shfl_xor(sv[v], 2, 32);
      sv[v] += __shfl_xor(sv[v], 4, 32);
      sv[v] += __shfl_xor(sv[v], 8, 32);
    }
    if (lo == 0) {
#pragma unroll
      for (int v = 0; v < 8; ++v) {
        const int row = rowBase + v + 8 * hi;
        if (row < N) s_out[row] = sv[v];
      }
    }
  }

  __syncthreads();
  fill_wlds(Wlds, W2, tid, 128);
  __syncthreads();

  if (active) {
    zero_acc(acc);
    wmma_row_gemm(X, mC, Wlds, hi, lo, acc);
#pragma unroll
    for (int nt = 0; nt < 8; ++nt) {
      const int col = nt * 16 + lo;
#pragma unroll
      for (int v = 0; v < 8; ++v) {
        const int row = rowBase + v + 8 * hi;
        if (row < N) P_out[(size_t)row * D + col] = acc[nt][v];
      }
    }
  }
}

// ---------------------------------------------------------------------------
// Stage 3: per-edge score + atomic segment-max.  One wave per edge (lane owns
// 4 of 128 dims, inner dim is DE=16).
// ---------------------------------------------------------------------------
__global__ __launch_bounds__(256) void edge_score_kernel(
    const float* __restrict__ EA, const int* __restrict__ EI, int E,
    const float* __restrict__ W_edge, const float* __restrict__ b_edge,
    const float* __restrict__ Wa_e, const float* __restrict__ b_attn,
    const float* __restrict__ s_src, const float* __restrict__ s_dst,
    float* __restrict__ scores, unsigned int* __restrict__ segmax) {
  __shared__ float We[DE * D];
  const int tid = threadIdx.x;
  for (int i = tid; i < DE * D; i += 256) We[i] = W_edge[i];
  __syncthreads();

  const int e = blockIdx.x * 8 + (tid >> 5);
  if (e >= E) return;
  const int lane = tid & 31;
  const float* ea = EA + (size_t)e * DE;
  float a[DE];
#pragma unroll
  for (int k = 0; k < DE; ++k) a[k] = ea[k];
  const int d0 = lane * 4;
  float s = 0.f;
#pragma unroll
  for (int j = 0; j < 4; ++j) {
    const int d = d0 + j;
    float acc = b_edge[d];
#pragma unroll
    for (int k = 0; k < DE; ++k) acc += a[k] * We[k * D + d];
    s += tanhf(acc) * Wa_e[d];
  }
  s += __shfl_xor(s, 16, 32);
  s += __shfl_xor(s, 8, 32);
  s += __shfl_xor(s, 4, 32);
  s += __shfl_xor(s, 2, 32);
  s += __shfl_xor(s, 1, 32);
  if (lane == 0) {
    const int src = EI[e];
    const int dst = EI[E + e];
    const float sc = s + s_src[src] + s_dst[dst] + b_attn[0];
    scores[e] = sc;
    atomicMax(&segmax[dst], encode_f32(sc));
  }
}

// ---------------------------------------------------------------------------
// Stage 4: ex = exp(score - segmax[dst]); seg_sum[dst] += ex (in-place).
// ---------------------------------------------------------------------------
__global__ __launch_bounds__(256) void edge_exp_kernel(
    const int* __restrict__ EI, int E, const unsigned int* __restrict__ segmax,
    float* __restrict__ scores, float* __restrict__ seg_sum) {
  const int e = blockIdx.x * 256 + threadIdx.x;
  if (e >= E) return;
  const int dst = EI[E + e];
  const float ex = expf(scores[e] - decode_f32(segmax[dst]));
  scores[e] = ex;
  unsafeAtomicAdd(&seg_sum[dst], ex);
}

// ---------------------------------------------------------------------------
// Stage 5: aggregate.  msg = SM[src] + edge_attr @ Wem + bmt, weighted by
// alpha = ex / seg_sum[dst], scatter-added into agg[dst].  One wave per edge.
// ---------------------------------------------------------------------------
__global__ __launch_bounds__(256) void edge_agg_kernel(
    const float* __restrict__ EA, const int* __restrict__ EI, int E,
    const float* __restrict__ Wem, const float* __restrict__ bmt,
    const float* __restrict__ SM, const float* __restrict__ ex,
    const float* __restrict__ seg_sum, float* __restrict__ agg) {
  __shared__ float We[DE * D];
  __shared__ float bl[D];
  const int tid = threadIdx.x;
  for (int i = tid; i < DE * D; i += 256) We[i] = Wem[i];
  if (tid < D) bl[tid] = bmt[tid];
  __syncthreads();

  const int e = blockIdx.x * 8 + (tid >> 5);
  if (e >= E) return;
  const int lane = tid & 31;
  const int src = EI[e];
  const int dst = EI[E + e];
  const float ss = seg_sum[dst];
  const float alpha = ex[e] / (ss > 0.f ? ss : 1.f);
  const float* ea = EA + (size_t)e * DE;
  float a[DE];
#pragma unroll
  for (int k = 0; k < DE; ++k) a[k] = ea[k];
  const int d0 = lane * 4;
#pragma unroll
  for (int j = 0; j < 4; ++j) {
    const int d = d0 + j;
    float m = SM[(size_t)src * D + d] + bl[d];
#pragma unroll
    for (int k = 0; k < DE; ++k) m += a[k] * We[k * D + d];
    unsafeAtomicAdd(&agg[(size_t)dst * D + d], alpha * m);
  }
}

// ---------------------------------------------------------------------------
// Stage 6: out = LayerNorm(dst_x + (dst_x@W_out_top) + agg@W_out_bot + b_out)
// U1 already holds dst_x@W_out_top.  WMMA GEMM + in-register LayerNorm.
// ---------------------------------------------------------------------------
__global__ __launch_bounds__(128) void final_kernel(
    const float* __restrict__ AGG, const float* __restrict__ DX,
    const float* __restrict__ U1, const float* __restrict__ Wob,
    const float* __restrict__ b_out, const float* __restrict__ gamma,
    const float* __restrict__ beta, float* __restrict__ out, int N) {
  __shared__ float Wlds[D * D];
  const int tid = threadIdx.x;
  const int wave = tid >> 5, lane = tid & 31, lo = lane & 15, hi = lane >> 4;
  const int rowBase = blockIdx.x * 64 + wave * 16;
  const int m = rowBase + lo;
  const int mC = (m < N) ? m : (N > 0 ? N - 1 : 0);

  fill_wlds(Wlds, Wob, tid, 128);
  __syncthreads();
  if (rowBase >= N) return;

  v8f acc[8];
  zero_acc(acc);
  wmma_row_gemm(AGG, mC, Wlds, hi, lo, acc);

  float h[8][8];
  float s1[8], s2[8];
#pragma unroll
  for (int v = 0; v < 8; ++v) { s1[v] = 0.f; s2[v] = 0.f; }
#pragma unroll
  for (int nt = 0; nt < 8; ++nt) {
    const int col = nt * 16 + lo;
    const float bo = b_out[col];
#pragma unroll
    for (int v = 0; v < 8; ++v) {
      const int row = rowBase + v + 8 * hi;
      const int rC = (row < N) ? row : (N - 1);
      const float val =
          acc[nt][v] + bo + DX[(size_t)rC * D + col] + U1[(size_t)rC * D + col];
      h[nt][v] = val;
      s1[v] += val;
      s2[v] += val * val;
    }
  }
#pragma unroll
  for (int v = 0; v < 8; ++v) {  // butterfly over 16-lane half-wave (row group)
    s1[v] += __shfl_xor(s1[v], 1, 32);
    s2[v] += __shfl_xor(s2[v], 1, 32);
    s1[v] += __shfl_xor(s1[v], 2, 32);
    s2[v] += __shfl_xor(s2[v], 2, 32);
    s1[v] += __shfl_xor(s1[v], 4, 32);
    s2[v] += __shfl_xor(s2[v], 4, 32);
    s1[v] += __shfl_xor(s1[v], 8, 32);
    s2[v] += __shfl_xor(s2[v], 8, 32);
  }
  float mean[8], inv[8];
#pragma unroll
  for (int v = 0; v < 8; ++v) {
    mean[v] = s1[v] * (1.0f / D);
    const float var = s2[v] * (1.0f / D) - mean[v] * mean[v];
    inv[v] = rsqrtf(var + LN_EPS);
  }
#pragma unroll
  for (int nt = 0; nt < 8; ++nt) {
    const int col = nt * 16 + lo;
    const float g = gamma[col], bb = beta[col];
#pragma unroll
    for (int v = 0; v < 8; ++v) {
      const int row = rowBase + v + 8 * hi;
      if (row < N)
        out[(size_t)row * D + col] = (h[nt][v] - mean[v]) * inv[v] * g + bb;
    }
  }
}

// ---------------------------------------------------------------------------
extern "C" void kernel_launch(void* const* d_in, const int* in_sizes, int n_in,
                              void* d_out, int out_size, void* d_ws,
                              size_t ws_size, hipStream_t stream) {
  const float* src_x     = (const float*)d_in[0];
  const float* dst_x     = (const float*)d_in[1];
  const float* edge_attr = (const float*)d_in[2];
  const int*   edge_index= (const int*)  d_in[3];
  const float* W_src  = (const float*)d_in[4];
  const float* b_src  = (const float*)d_in[5];
  const float* W_dst  = (const float*)d_in[6];
  const float* b_dst  = (const float*)d_in[7];
  const float* W_edge = (const float*)d_in[8];
  const float* b_edge = (const float*)d_in[9];
  const float* W_attn = (const float*)d_in[10];
  const float* b_attn = (const float*)d_in[11];
  const float* W_msg  = (const float*)d_in[12];
  const float* b_msg  = (const float*)d_in[13];
  const float* W_out  = (const float*)d_in[14];
  const float* b_out  = (const float*)d_in[15];
  const float* ln_g   = (const float*)d_in[16];
  const float* ln_b   = (const float*)d_in[17];

  const int N_SRC = in_sizes[0] / D;
  const int N_DST = in_sizes[1] / D;
  const int E     = in_sizes[2] / DE;

  // workspace layout (zeroed region first: segmax, seg_sum, agg)
  float* p = (float*)d_ws;
  unsigned int* segmax = (unsigned int*)p;  p += N_DST;
  float* seg_sum = p;                       p += N_DST;
  float* agg = p;                           p += (size_t)N_DST * D;
  float* Wsm = p;                           p += D * D;
  float* Wem = p;                           p += DE * D;
  float* bmt = p;                           p += D;
  float* s_src = p;                         p += N_SRC;
  float* s_dst = p;                         p += N_DST;
  float* SM = p;                            p += (size_t)N_SRC * D;
  float* U1 = p;                            p += (size_t)N_DST * D;
  float* scores = p;                        p += E;

  const size_t zeroBytes = ((size_t)N_DST * 2 + (size_t)N_DST * D) * sizeof(float);
  hipMemsetAsync((void*)segmax, 0, zeroBytes, stream);

  fuse_weights_kernel<<<D + DE + 1, D, 0, stream>>>(
      W_src, W_msg, W_edge, b_src, b_edge, b_msg, Wsm, Wem, bmt);

  const int nsb = (N_SRC + 63) / 64;
  const int ndb = (N_DST + 63) / 64;
  node_stage_kernel<<<nsb, 128, 0, stream>>>(src_x, N_SRC, W_src, b_src,
                                             W_attn, s_src, Wsm, SM);
  node_stage_kernel<<<ndb, 128, 0, stream>>>(dst_x, N_DST, W_dst, b_dst,
                                             W_attn + D, s_dst, W_out, U1);
  edge_score_kernel<<<(E + 7) / 8, 256, 0, stream>>>(
      edge_attr, edge_index, E, W_edge, b_edge, W_attn + 2 * D, b_attn, s_src,
      s_dst, scores, segmax);
  edge_exp_kernel<<<(E + 255) / 256, 256, 0, stream>>>(edge_index, E, segmax,
                                                       scores, seg_sum);
  edge_agg_kernel<<<(E + 7) / 8, 256, 0, stream>>>(
      edge_attr, edge_index, E, Wem, bmt, SM, scores, seg_sum, agg);
  final_kernel<<<ndb, 128, 0, stream>>>(agg, dst_x, U1, W_out + D * D, b_out,
                                        ln_g, ln_b, (float*)d_out, N_DST);
}